// ConvNet_11450382811698
// MI455X (gfx1250) — compile-verified
//
#include <hip/hip_runtime.h>

// ---------------------------------------------------------------------------
// Shapes (fixed by the reference)
// ---------------------------------------------------------------------------
#define Bsz 32
#define Lsz 256
#define Csz 16
#define Dsz 128
#define KTOT 384            // 3 taps * 128 din  (im2col K)
#define NKCHUNK 96          // KTOT / 4   (WMMA f32 K=4)
#define WAVES 8             // 256-thread block = 8 wave32
#define WT2_PITCH 288       // floats per kg-pair row; 288 % 64 == 32 -> halves use disjoint banks
#define WT2_FLOATS (192 * WT2_PITCH)   // 55296 floats = 216 KB
#define XT_FLOATS (64 * 36)            // 2304 floats per wave (64 din pairs x 18 cols x 2)
#define FEATS_FLOATS (Dsz * 16)        // char path per-block output staging

typedef __attribute__((ext_vector_type(2))) float v2f;
typedef __attribute__((ext_vector_type(8))) float v8f;

// ---------------------------------------------------------------------------
// Stage im2col'ed conv weight into LDS:
//   Wt[kg = tap*128 + din][dout],  pair-interleaved:
//   lds[(kg>>1)*WT2_PITCH + dout*2 + (kg&1)]
// Global reads are coalesced (consecutive threads -> consecutive din, 12B apart).
// ---------------------------------------------------------------------------
__device__ __forceinline__ void load_wt(const float* __restrict__ W, float* lds, int tid) {
  for (int idx = tid; idx < Dsz * Dsz; idx += 256) {
    const int dout = idx >> 7;
    const int din  = idx & 127;
    const float* wp = W + dout * 384 + din * 3;
    const float w0 = wp[0], w1 = wp[1], w2 = wp[2];
    const int kg0 = din;          // tap 0
    const int kg1 = 128 + din;    // tap 1
    const int kg2 = 256 + din;    // tap 2
    lds[(kg0 >> 1) * WT2_PITCH + dout * 2 + (kg0 & 1)] = w0;
    lds[(kg1 >> 1) * WT2_PITCH + dout * 2 + (kg1 & 1)] = w1;
    lds[(kg2 >> 1) * WT2_PITCH + dout * 2 + (kg2 & 1)] = w2;
  }
}

// ---------------------------------------------------------------------------
// Fragment fetch for one K-chunk (K=4):
// A fragment (16x4):  lane(r,h): v0 = A[M=r, K=base+2h], v1 = A[M=r, K=base+2h+1]
// B fragment (4x16):  lane(r,h): v0 = B[K=base+2h, N=r], v1 = B[K=base+2h+1, N=r]
// xt layout: element (din, cc) at (din>>1)*36 + cc*2 + (din&1); cc = col + tap
// (columns 0 and 17 are the conv halo).
// ---------------------------------------------------------------------------
__device__ __forceinline__ void load_frags(const float* __restrict__ wt,
                                           const float* __restrict__ xt,
                                           int kk, int r, int h,
                                           v2f a[8], v2f* bf) {
  const int kg0  = kk * 4 + 2 * h;      // even; (kg0, kg0+1) share the same tap
  const int tap  = kg0 >> 7;
  const int din0 = kg0 & 127;
  *bf = *(const v2f*)(xt + (din0 >> 1) * 36 + (r + tap) * 2);
  const float* wrow = wt + (kg0 >> 1) * WT2_PITCH + r * 2;
#pragma unroll
  for (int t = 0; t < 8; ++t) {
    a[t] = *(const v2f*)(wrow + t * 32);
  }
}

__device__ __forceinline__ void wmma_group(const v2f a[8], v2f bf, v8f acc[8]) {
#pragma unroll
  for (int t = 0; t < 8; ++t) {
    acc[t] = __builtin_amdgcn_wmma_f32_16x16x4_f32(
        /*neg_a=*/false, a[t], /*neg_b=*/false, bf,
        /*c_mod=*/(short)0, acc[t], /*reuse_a=*/false, /*reuse_b=*/false);
  }
}

// ---------------------------------------------------------------------------
// Core: 8 dout tiles (128 rows) x 16 cols, K = 384, f32 WMMA.
// Software-pipelined: double-buffered fragments so the 9-load group for the
// next K-chunk is in flight while the current chunk's 8 WMMAs issue.
// ---------------------------------------------------------------------------
__device__ __forceinline__ void conv_tiles(const float* __restrict__ wt,
                                           const float* __restrict__ xt,
                                           int r, int h, v8f acc[8]) {
  v2f a0[8], a1[8], b0, b1;
  load_frags(wt, xt, 0, r, h, a0, &b0);
#pragma unroll 1
  for (int kk = 0; kk < NKCHUNK - 2; kk += 2) {
    load_frags(wt, xt, kk + 1, r, h, a1, &b1);
    wmma_group(a0, b0, acc);
    load_frags(wt, xt, kk + 2, r, h, a0, &b0);
    wmma_group(a1, b1, acc);
  }
  // tail: chunks 94, 95
  load_frags(wt, xt, NKCHUNK - 1, r, h, a1, &b1);
  wmma_group(a0, b0, acc);
  wmma_group(a1, b1, acc);
}

// ---------------------------------------------------------------------------
// Char path: grid = 512 blocks x 256 threads; block handles 16 consecutive
// words (same b); each wave computes 2 words end-to-end.
// out[1][b][d][l] = bias[d] + max_c conv(chr_emb)[d][c]
// ---------------------------------------------------------------------------
__global__ __launch_bounds__(256) void char_conv_kernel(
    const int* __restrict__ words_in_char,
    const float* __restrict__ chr_table,
    const float* __restrict__ conv_w,
    const float* __restrict__ conv_b,
    float* __restrict__ out) {
  extern __shared__ float smem[];
  float* wt = smem;
  const int tid  = threadIdx.x;
  const int lane = tid & 31;
  const int w    = tid >> 5;
  float* xt    = smem + WT2_FLOATS + w * XT_FLOATS;
  float* feats = smem + WT2_FLOATS + WAVES * XT_FLOATS;   // [128][16]

  load_wt(conv_w, wt, tid);
  __syncthreads();

  const int gw0 = blockIdx.x * 16;   // first flat word index of the block
  const int b   = gw0 >> 8;
  const int l0  = gw0 & 255;
  const int r   = lane & 15;
  const int h   = lane >> 4;

  for (int wi = 0; wi < 2; ++wi) {
    const int lw = w * 2 + wi;               // 0..15 within block
    const int g  = gw0 + lw;
    const int* wic = words_in_char + g * Csz;

    // Gather char embeddings into xt (cols 1..16), zero halo (cols 0, 17).
    for (int cc = 0; cc < 18; ++cc) {
      const float* src = nullptr;
      if (cc >= 1 && cc <= 16) {
        const int ch = wic[cc - 1];
        src = chr_table + ch * Dsz;
      }
      for (int d = lane; d < Dsz; d += 32) {
        xt[(d >> 1) * 36 + cc * 2 + (d & 1)] = src ? src[d] : 0.0f;
      }
    }

    v8f acc[8] = {};
    conv_tiles(wt, xt, r, h, acc);

    // Max over the 16 conv positions (N dim = 16 lanes within each half).
#pragma unroll
    for (int t = 0; t < 8; ++t) {
#pragma unroll
      for (int j = 0; j < 8; ++j) {
        float v = acc[t][j];
        v = fmaxf(v, __shfl_xor(v, 1, 32));
        v = fmaxf(v, __shfl_xor(v, 2, 32));
        v = fmaxf(v, __shfl_xor(v, 4, 32));
        v = fmaxf(v, __shfl_xor(v, 8, 32));
        if (r == j) {                          // one writer per half
          feats[(t * 16 + 8 * h + j) * 16 + lw] = v;
        }
      }
    }
  }
  __syncthreads();

  // Coalesced store: out[1][b][d][l0..l0+15], add bias here.
  for (int i = tid; i < Dsz * 16; i += 256) {
    const int d = i >> 4;
    const int j = i & 15;
    out[((Bsz + b) * Dsz + d) * Lsz + l0 + j] = feats[d * 16 + j] + conv_b[d];
  }
}

// ---------------------------------------------------------------------------
// Word path: grid = 64 blocks x 256 threads; wave handles one (b, l-tile of 16).
// out[0][b][d][l] = bias[d] + conv(word_emb)[d][l]
// ---------------------------------------------------------------------------
__global__ __launch_bounds__(256) void word_conv_kernel(
    const int* __restrict__ word_vector,
    const float* __restrict__ word_table,
    const float* __restrict__ conv_w,
    const float* __restrict__ conv_b,
    float* __restrict__ out) {
  extern __shared__ float smem[];
  float* wt = smem;
  const int tid  = threadIdx.x;
  const int lane = tid & 31;
  const int w    = tid >> 5;
  float* xt = smem + WT2_FLOATS + w * XT_FLOATS;

  load_wt(conv_w, wt, tid);
  __syncthreads();

  const int task = blockIdx.x * WAVES + w;   // 0..511 = 32 b * 16 tiles
  const int b    = task >> 4;
  const int l0   = (task & 15) * 16;
  const int r    = lane & 15;
  const int h    = lane >> 4;

  // Gather word embeddings for l0-1 .. l0+16 (18 cols, zero outside [0,L)).
  for (int cc = 0; cc < 18; ++cc) {
    const int l = l0 + cc - 1;
    const float* src = nullptr;
    if (l >= 0 && l < Lsz) {
      const int widx = word_vector[b * Lsz + l];
      src = word_table + widx * Dsz;
    }
    for (int d = lane; d < Dsz; d += 32) {
      xt[(d >> 1) * 36 + cc * 2 + (d & 1)] = src ? src[d] : 0.0f;
    }
  }

  v8f acc[8] = {};
  conv_tiles(wt, xt, r, h, acc);

  // Direct coalesced stores: row d = t*16 + 8h + j, col l = l0 + r.
#pragma unroll
  for (int t = 0; t < 8; ++t) {
#pragma unroll
    for (int j = 0; j < 8; ++j) {
      const int row = t * 16 + 8 * h + j;
      out[(b * Dsz + row) * Lsz + l0 + r] = acc[t][j] + conv_b[row];
    }
  }
}

// ---------------------------------------------------------------------------
// Launch
// Inputs (setup_inputs order):
//   0 word_vector   int   [B,L]
//   1 words_in_char int   [B,L,C]
//   2 word_table    f32   [50000,D]
//   3 chr_table     f32   [128,D]
//   4 conv_chr_w    f32   [D,D,3]
//   5 conv_chr_b    f32   [D]
//   6 conv_word_w   f32   [D,D,3]
//   7 conv_word_b   f32   [D]
// Output: f32 [2,B,D,L]  (word conv at [0], char feats at [1])
// ---------------------------------------------------------------------------
extern "C" void kernel_launch(void* const* d_in, const int* in_sizes, int n_in,
                              void* d_out, int out_size, void* d_ws, size_t ws_size,
                              hipStream_t stream) {
  (void)in_sizes; (void)n_in; (void)out_size; (void)d_ws; (void)ws_size;
  const int*   word_vector   = (const int*)d_in[0];
  const int*   words_in_char = (const int*)d_in[1];
  const float* word_table    = (const float*)d_in[2];
  const float* chr_table     = (const float*)d_in[3];
  const float* conv_chr_w    = (const float*)d_in[4];
  const float* conv_chr_b    = (const float*)d_in[5];
  const float* conv_word_w   = (const float*)d_in[6];
  const float* conv_word_b   = (const float*)d_in[7];
  float* out = (float*)d_out;

  const size_t char_lds = (size_t)(WT2_FLOATS + WAVES * XT_FLOATS + FEATS_FLOATS) * sizeof(float);
  const size_t word_lds = (size_t)(WT2_FLOATS + WAVES * XT_FLOATS) * sizeof(float);

  static bool attr_done = false;
  if (!attr_done) {   // host-side attribute set; not a stream op (safe under capture)
    hipFuncSetAttribute((const void*)char_conv_kernel,
                        hipFuncAttributeMaxDynamicSharedMemorySize, (int)char_lds);
    hipFuncSetAttribute((const void*)word_conv_kernel,
                        hipFuncAttributeMaxDynamicSharedMemorySize, (int)word_lds);
    attr_done = true;
  }

  char_conv_kernel<<<dim3(Bsz * Lsz / 16), dim3(256), char_lds, stream>>>(
      words_in_char, chr_table, conv_chr_w, conv_chr_b, out);
  word_conv_kernel<<<dim3(Bsz * 16 / WAVES), dim3(256), word_lds, stream>>>(
      word_vector, word_table, conv_word_w, conv_word_b, out);
}